// DifferentiableBSpline_32865089749328
// MI455X (gfx1250) — compile-verified
//
#include <hip/hip_runtime.h>

typedef __attribute__((ext_vector_type(2))) float v2f;
typedef __attribute__((ext_vector_type(8))) float v8f;

#define NCP  16
#define NOUT 256

// ---------------------------------------------------------------------------
// Kernel 1: build the constant spline weight matrix W[256][16] into workspace.
// seg = min(j/17, 14), t = (j - 17*seg)/17  (replicates _spline_schedule for
// n_cp=16, num_output_points=256). Hermite basis folded with the mirrored
// endpoint extension into weights on the 16 real control points.
// ---------------------------------------------------------------------------
__global__ void spline_weights_kernel(float* __restrict__ W) {
    int j = blockIdx.x * blockDim.x + threadIdx.x;
    if (j >= NOUT) return;
    int seg = j / 17;
    if (seg > 14) seg = 14;
    float t  = (float)(j - 17 * seg) * (1.0f / 17.0f);
    float t2 = t * t, t3 = t2 * t;
    float h00 =  2.f * t3 - 3.f * t2 + 1.f;
    float h10 =        t3 - 2.f * t2 + t;
    float h01 = -2.f * t3 + 3.f * t2;
    float h11 =        t3 -       t2;
    // out = h00*p0 + h10*m0 + h01*p1 + h11*m1, m = 0.5*(next - prev)
    // => weights on ext[seg..seg+3]:
    float w[4] = { -0.5f * h10,
                    h00 - 0.5f * h11,
                    h01 + 0.5f * h10,
                    0.5f * h11 };
    float row[NCP];
#pragma unroll
    for (int k = 0; k < NCP; ++k) row[k] = 0.f;
#pragma unroll
    for (int q = 0; q < 4; ++q) {
        int   e = seg + q;     // index into 18-long extended array
        float c = w[q];
        if (e == 0)       { row[0]  += 2.f * c; row[1]  -= c; }  // ext0 = 2cp0 - cp1
        else if (e == 17) { row[15] += 2.f * c; row[14] -= c; }  // ext17 = 2cp15 - cp14
        else              { row[e - 1] += c; }
    }
#pragma unroll
    for (int k = 0; k < NCP; ++k) W[j * NCP + k] = row[k];
}

// ---------------------------------------------------------------------------
// Kernel 2: batched GEMM  out[b](256x2) = W(256x16) @ cp[b](16x2)
// via V_WMMA_F32_16X16X4_F32 chains (K = 16 -> 4 chained WMMAs).
//
// Wave w of a 256-thread block owns j-tiles {w, w+8} (16 output rows each).
// A-fragments (shared W) are loop-invariant per wave. Columns of each WMMA
// are 16 batches; x and y channels use separate accumulator chains so every
// lane can emit interleaved (x,y) pairs as 8-byte stores.
// ---------------------------------------------------------------------------
__global__ __launch_bounds__(256) void spline_eval_kernel(
    const float* __restrict__ cp,   // [B][16][2]
    const float* __restrict__ W,    // [256][16]
    float* __restrict__ out,        // [B][256][2]
    int groups_per_block,
    long long Btot)
{
    const int lane = threadIdx.x & 31;
    const int wave = threadIdx.x >> 5;   // 0..7
    const int half = lane >> 4;          // 0/1 : K-half selector per ISA layout
    const int l15  = lane & 15;

    // ---- A fragments: A[tile][chunk], 16x4 f32 each -----------------------
    // lane L holds A[M = L&15, K = 2*half + {0,1}] (VGPR0/VGPR1)
    v2f A[2][4];
#pragma unroll
    for (int ts = 0; ts < 2; ++ts) {
        const int j = (wave + 8 * ts) * 16 + l15;
        const float* wrow = W + j * NCP;
#pragma unroll
        for (int kc = 0; kc < 4; ++kc) {
            const int k = kc * 4 + 2 * half;
            v2f a; a.x = wrow[k]; a.y = wrow[k + 1];
            A[ts][kc] = a;
        }
    }

    const long long b0base = (long long)blockIdx.x * groups_per_block * 16;

    for (int g = 0; g < groups_per_block; ++g) {
        const long long b0 = b0base + (long long)g * 16;
        if (b0 >= Btot) break;                       // uniform guard

        // ---- B fragments: 4x16 f32 per K-chunk, one per channel ----------
        // lane L = column n = batch b0 + (L&15); holds B[K = 2*half+{0,1}, n]
        long long bb = b0 + l15;
        if (bb >= Btot) bb = Btot - 1;               // clamp tail loads
        const float* cpb = cp + bb * (NCP * 2);
        v2f Bx[4], By[4];
#pragma unroll
        for (int kc = 0; kc < 4; ++kc) {
            const int k = kc * 4 + 2 * half;
            const float* p = cpb + k * 2;            // [k][ch]
            v2f bx, by;
            bx.x = p[0]; by.x = p[1];                // K = k
            bx.y = p[2]; by.y = p[3];                // K = k+1
            Bx[kc] = bx; By[kc] = by;
        }

        const bool colok = (b0 + l15) < Btot;

#pragma unroll
        for (int ts = 0; ts < 2; ++ts) {
            v8f cx = {};
            v8f cy = {};
#pragma unroll
            for (int kc = 0; kc < 4; ++kc) {
                cx = __builtin_amdgcn_wmma_f32_16x16x4_f32(
                        false, A[ts][kc], false, Bx[kc], (short)0, cx, false, false);
                cy = __builtin_amdgcn_wmma_f32_16x16x4_f32(
                        false, A[ts][kc], false, By[kc], (short)0, cy, false, false);
            }
            // D layout: VGPR r, lane L -> D[r + 8*half, L&15]
            // row -> j = (wave+8*ts)*16 + r + 8*half ; col -> batch b0 + l15
            if (colok) {
                const int j0 = (wave + 8 * ts) * 16 + 8 * half;
                float* obase = out + ((b0 + l15) * NOUT + j0) * 2;
#pragma unroll
                for (int r = 0; r < 8; ++r) {
                    v2f d; d.x = cx[r]; d.y = cy[r];
                    *(v2f*)(obase + r * 2) = d;      // 8B (x,y) -> global_store_b64
                }
            }
        }
    }
}

// ---------------------------------------------------------------------------
extern "C" void kernel_launch(void* const* d_in, const int* in_sizes, int n_in,
                              void* d_out, int out_size, void* d_ws, size_t ws_size,
                              hipStream_t stream) {
    const float* cp  = (const float*)d_in[0];          // [B][16][2] f32
    // d_in[1] = num_output_points (always 256; schedule baked in)
    float* W   = (float*)d_ws;                          // 256*16 f32 = 16 KB
    float* out = (float*)d_out;                         // [B][256][2] f32
    const long long B = (long long)in_sizes[0] / (NCP * 2);

    spline_weights_kernel<<<1, NOUT, 0, stream>>>(W);

    const int gpb = 4;                                  // 16-batch groups per block
    const long long bpb = 16LL * gpb;                   // 64 batches per block
    const int grid = (int)((B + bpb - 1) / bpb);        // 1024 blocks for B=65536
    spline_eval_kernel<<<grid, 256, 0, stream>>>(cp, W, out, gpb, B);
}